// Tip3pTorchFF_46969762349286
// MI455X (gfx1250) — compile-verified
//
#include <hip/hip_runtime.h>
#include <math.h>

#define ALPHA_F   3.0f
#define CUT2_F    0.81f            /* 0.9^2 */
#define PREF_F    138.935456f
#define MAX_HKL   8
#define KGRID     17               /* 2*MAX_HKL+1 */
#define NK        (KGRID*KGRID*KGRID - 1)                 /* 4912 */
#define KCENTER   (MAX_HKL*KGRID*KGRID + MAX_HKL*KGRID + MAX_HKL)  /* 2456 */
#define NEXCL_MAX 8
#define TWO_PI_F  6.28318530717958647692f

typedef float v2f __attribute__((ext_vector_type(2)));
typedef float v8f __attribute__((ext_vector_type(8)));

static __device__ __forceinline__ float block_reduce_add(float v, float* red) {
  int t = threadIdx.x;
  red[t] = v;
  __syncthreads();
  for (int s = blockDim.x >> 1; s > 0; s >>= 1) {
    if (t < s) red[t] += red[t + s];
    __syncthreads();
  }
  return red[0];
}

/* ---------------- init: zero the energy accumulator + exclusion counts ----- */
__global__ void init_kernel(float* out, int* counts, int N) {
  int t = blockIdx.x * blockDim.x + threadIdx.x;
  if (t == 0) out[0] = 0.0f;
  if (t < N) counts[t] = 0;
}

/* ---------------- build per-atom exclusion lists --------------------------- */
__global__ void build_excl_kernel(const int* pairs, int np, int* counts,
                                  int* lists, int N) {
  int t = blockIdx.x * blockDim.x + threadIdx.x;
  if (t >= np) return;
  int a = pairs[2 * t + 0];
  int b = pairs[2 * t + 1];
  if (a < 0 || a >= N || b < 0 || b >= N) return;
  int ia = atomicAdd(&counts[a], 1);
  if (ia < NEXCL_MAX) lists[a * NEXCL_MAX + ia] = b;
  int ib = atomicAdd(&counts[b], 1);
  if (ib < NEXCL_MAX) lists[b * NEXCL_MAX + ib] = a;
}

/* ---------------- bonds + angles + erf-exclusion + self -------------------- */
__global__ __launch_bounds__(256)
void small_kernel(const float* coords, const float* box,
                  const int* bonds, const float* b0, const float* kb, int nb,
                  const int* angles, const float* th0, const float* kth, int na,
                  const int* cexcl, int nx,
                  const float* charges, int N, float* out) {
  __shared__ float red[256];
  int t = blockIdx.x * blockDim.x + threadIdx.x;
  float bx = box[0], by = box[1], bz = box[2];
  float ibx = 1.0f / bx, iby = 1.0f / by, ibz = 1.0f / bz;
  float acc = 0.0f;

  if (t < nb) {  /* harmonic bonds */
    int a = bonds[2 * t + 0], b = bonds[2 * t + 1];
    float dx = coords[3 * a + 0] - coords[3 * b + 0]; dx -= bx * rintf(dx * ibx);
    float dy = coords[3 * a + 1] - coords[3 * b + 1]; dy -= by * rintf(dy * iby);
    float dz = coords[3 * a + 2] - coords[3 * b + 2]; dz -= bz * rintf(dz * ibz);
    float r = sqrtf(dx * dx + dy * dy + dz * dz);
    float d = r - b0[t];
    acc += 0.5f * kb[t] * d * d;
  }
  if (t < na) {  /* harmonic angles */
    int a0 = angles[3 * t + 0], a1 = angles[3 * t + 1], a2 = angles[3 * t + 2];
    float v1x = coords[3 * a0 + 0] - coords[3 * a1 + 0]; v1x -= bx * rintf(v1x * ibx);
    float v1y = coords[3 * a0 + 1] - coords[3 * a1 + 1]; v1y -= by * rintf(v1y * iby);
    float v1z = coords[3 * a0 + 2] - coords[3 * a1 + 2]; v1z -= bz * rintf(v1z * ibz);
    float v2x = coords[3 * a2 + 0] - coords[3 * a1 + 0]; v2x -= bx * rintf(v2x * ibx);
    float v2y = coords[3 * a2 + 1] - coords[3 * a1 + 1]; v2y -= by * rintf(v2y * iby);
    float v2z = coords[3 * a2 + 2] - coords[3 * a1 + 2]; v2z -= bz * rintf(v2z * ibz);
    float n1 = sqrtf(v1x * v1x + v1y * v1y + v1z * v1z);
    float n2 = sqrtf(v2x * v2x + v2y * v2y + v2z * v2z);
    float c = (v1x * v2x + v1y * v2y + v1z * v2z) / (n1 * n2);
    c = fminf(fmaxf(c, -1.0f + 1e-7f), 1.0f - 1e-7f);
    float d = acosf(c) - th0[t];
    acc += 0.5f * kth[t] * d * d;
  }
  if (t < nx) {  /* Ewald excluded-pair correction */
    int a = cexcl[2 * t + 0], b = cexcl[2 * t + 1];
    float dx = coords[3 * a + 0] - coords[3 * b + 0]; dx -= bx * rintf(dx * ibx);
    float dy = coords[3 * a + 1] - coords[3 * b + 1]; dy -= by * rintf(dy * iby);
    float dz = coords[3 * a + 2] - coords[3 * b + 2]; dz -= bz * rintf(dz * ibz);
    float r = sqrtf(dx * dx + dy * dy + dz * dz);
    float qx = charges[a] * charges[b];
    acc -= PREF_F * qx * erff(ALPHA_F * r) / r;
  }
  if (t < N) {  /* Ewald self energy */
    float q = charges[t];
    acc -= PREF_F * (ALPHA_F * 0.5641895835477563f) * q * q; /* alpha/sqrt(pi) */
  }

  float tot = block_reduce_add(acc, red);
  if (threadIdx.x == 0) atomicAdd(out, tot);
}

/* ---------------- real-space N^2: LJ + erfc Coulomb ------------------------ */
#define PTILE 256
__global__ __launch_bounds__(PTILE)
void pair_kernel(const float* coords, const float* charges, const int* types,
                 const float* sigma, const float* epsilon, const float* box,
                 const int* excnt, const int* exlist, int N, float* out) {
  __shared__ float4 sj[PTILE];
  __shared__ int    stj[PTILE];
  __shared__ float  red[PTILE];

  int i = blockIdx.x * PTILE + threadIdx.x;
  float xi = 0.f, yi = 0.f, zi = 0.f, qi = 0.f;
  int ti = 0;
  int ex[NEXCL_MAX];
  if (i < N) {
    xi = coords[3 * i + 0]; yi = coords[3 * i + 1]; zi = coords[3 * i + 2];
    qi = charges[i];
    ti = types[i];
    int nex = excnt[i]; if (nex > NEXCL_MAX) nex = NEXCL_MAX;
    #pragma unroll
    for (int e = 0; e < NEXCL_MAX; ++e) ex[e] = (e < nex) ? exlist[i * NEXCL_MAX + e] : -1;
  } else {
    #pragma unroll
    for (int e = 0; e < NEXCL_MAX; ++e) ex[e] = -1;
  }
  float bx = box[0], by = box[1], bz = box[2];
  float ibx = 1.0f / bx, iby = 1.0f / by, ibz = 1.0f / bz;
  float s00 = sigma[0], s01 = sigma[1], s10 = sigma[2], s11 = sigma[3];
  float e00 = epsilon[0], e01 = epsilon[1], e10 = epsilon[2], e11 = epsilon[3];

  float acc = 0.0f;
  for (int j0 = 0; j0 < N; j0 += PTILE) {
    int jj = j0 + threadIdx.x;
    if (jj < N) {
      sj[threadIdx.x]  = make_float4(coords[3 * jj + 0], coords[3 * jj + 1],
                                     coords[3 * jj + 2], charges[jj]);
      stj[threadIdx.x] = types[jj];
    }
    __syncthreads();
    int jmax = N - j0; if (jmax > PTILE) jmax = PTILE;
    if (i < N) {
      for (int jt = 0; jt < jmax; ++jt) {
        int j = j0 + jt;
        float4 pj = sj[jt];
        float dx = xi - pj.x; dx -= bx * rintf(dx * ibx);
        float dy = yi - pj.y; dy -= by * rintf(dy * iby);
        float dz = zi - pj.z; dz -= bz * rintf(dz * ibz);
        float r2 = dx * dx + dy * dy + dz * dz;
        if (r2 < CUT2_F && r2 > 1e-12f) {
          bool excl = false;
          #pragma unroll
          for (int e = 0; e < NEXCL_MAX; ++e) excl |= (ex[e] == j);
          if (!excl) {
            int tj = stj[jt];
            float sg = (ti == 0) ? (tj == 0 ? s00 : s01) : (tj == 0 ? s10 : s11);
            float ep = (ti == 0) ? (tj == 0 ? e00 : e01) : (tj == 0 ? e10 : e11);
            float ir2 = 1.0f / r2;
            float sr2 = sg * sg * ir2;
            float s6 = sr2 * sr2 * sr2;
            float elj = 4.0f * ep * (s6 * s6 - s6);
            float r = sqrtf(r2);
            float ec = qi * pj.w * erfcf(ALPHA_F * r) / r;
            acc += elj + PREF_F * ec;
          }
        }
      }
    }
    __syncthreads();
  }
  float tot = block_reduce_add(0.5f * acc, red);  /* pairs counted twice */
  if (threadIdx.x == 0) atomicAdd(out, tot);
}

/* ---------------- reciprocal-space Ewald via V_WMMA_F32_16X16X4_F32 -------- */
/* One wave per 16 k-vectors. A = 16 atoms x (x,y,z,0); B = (kx,ky,kz,0)^T x  */
/* 16 k-vectors; D[m][n] = phase = r_m . k_n per ISA 7.12.2 f32 layouts.      */
/* Full atom tiles are branch-free: b96 coords load + 2x b128 charge loads +  */
/* one WMMA + 8 sincos per lane. Only the remainder tile uses clamped loads   */
/* with value-selects (no EXEC divergence around the WMMA).                   */
__global__ __launch_bounds__(32)
void recip_kernel(const float* coords, const float* charges, const float* box,
                  int N, float* out) {
  const int lane = threadIdx.x;          /* wave32 */
  const int n    = lane & 15;            /* k-column in tile */
  const int kidx = blockIdx.x * 16 + n;
  int f = kidx + (kidx >= KCENTER ? 1 : 0);   /* skip (0,0,0) */
  if (kidx >= NK) f = KCENTER;                /* degenerate pad (filtered below) */
  const int h = f / (KGRID * KGRID) - MAX_HKL;
  const int k = (f / KGRID) % KGRID    - MAX_HKL;
  const int l = f % KGRID              - MAX_HKL;

  const float bx = box[0], by = box[1], bz = box[2];
  const float kvx = TWO_PI_F * (float)h / bx;
  const float kvy = TWO_PI_F * (float)k / by;
  const float kvz = TWO_PI_F * (float)l / bz;

  const bool lowhalf = (lane < 16);
  /* B matrix (4x16 f32): lanes 0-15 hold K=0,1 ; lanes 16-31 hold K=2,3 */
  v2f bmat;
  bmat.x = lowhalf ? kvx : kvz;
  bmat.y = lowhalf ? kvy : 0.0f;

  float sre = 0.0f, sim = 0.0f;
  const int rbase = (lane & 16) >> 1;    /* D rows: 0..7 (low half) or 8..15 */
  const int nfull = N >> 4;              /* full 16-atom tiles */

  for (int tile = 0; tile < nfull; ++tile) {
    const int mbase = tile << 4;
    const int m = mbase + n;
    const float* cp = coords + 3 * m;    /* unconditional b96 */
    float cx = cp[0], cy = cp[1], cz = cp[2];
    v2f amat;                            /* A (16x4 f32) per ISA layout */
    amat.x = lowhalf ? cx : cz;
    amat.y = lowhalf ? cy : 0.0f;

    v8f ph;
#if defined(__HIP_DEVICE_COMPILE__) && __has_builtin(__builtin_amdgcn_wmma_f32_16x16x4_f32)
    v8f czero = {0.f, 0.f, 0.f, 0.f, 0.f, 0.f, 0.f, 0.f};
    ph = __builtin_amdgcn_wmma_f32_16x16x4_f32(false, amat, false, bmat,
                                               (short)0, czero, false, false);
#else
    #pragma unroll
    for (int r = 0; r < 8; ++r) {
      const float* cr = coords + 3 * (mbase + rbase + r);
      ph[r] = kvx * cr[0] + kvy * cr[1] + kvz * cr[2];
    }
#endif
    /* contiguous 32B of charges per lane -> two aligned b128 loads */
    const float4 qa = *(const float4*)(charges + mbase + rbase);
    const float4 qb = *(const float4*)(charges + mbase + rbase + 4);
    const float qarr[8] = {qa.x, qa.y, qa.z, qa.w, qb.x, qb.y, qb.z, qb.w};
    #pragma unroll
    for (int r = 0; r < 8; ++r) {
      float s, c;
      __sincosf(ph[r], &s, &c);
      sre += qarr[r] * c;
      sim += qarr[r] * s;
    }
  }

  if (N & 15) {  /* remainder tile: clamped loads, value-masked charges */
    const int mbase = nfull << 4;
    int m  = mbase + n;
    int mc = (m < N) ? m : (N - 1);
    const float* cp = coords + 3 * mc;
    float cx = cp[0], cy = cp[1], cz = cp[2];
    v2f amat;
    amat.x = lowhalf ? cx : cz;
    amat.y = lowhalf ? cy : 0.0f;

    v8f ph;
#if defined(__HIP_DEVICE_COMPILE__) && __has_builtin(__builtin_amdgcn_wmma_f32_16x16x4_f32)
    v8f czero = {0.f, 0.f, 0.f, 0.f, 0.f, 0.f, 0.f, 0.f};
    ph = __builtin_amdgcn_wmma_f32_16x16x4_f32(false, amat, false, bmat,
                                               (short)0, czero, false, false);
#else
    #pragma unroll
    for (int r = 0; r < 8; ++r) {
      int mr = mbase + rbase + r;
      const float* cr = coords + 3 * ((mr < N) ? mr : (N - 1));
      ph[r] = kvx * cr[0] + kvy * cr[1] + kvz * cr[2];
    }
#endif
    #pragma unroll
    for (int r = 0; r < 8; ++r) {
      int mq = mbase + rbase + r;
      float q = charges[(mq < N) ? mq : (N - 1)];
      q = (mq < N) ? q : 0.0f;           /* select on value, not on load */
      float s, c;
      __sincosf(ph[r], &s, &c);
      sre += q * c;
      sim += q * s;
    }
  }

  /* fold the two row-halves (lane L and L+16 share the same k-column) */
  sre += __shfl_xor(sre, 16, 32);
  sim += __shfl_xor(sim, 16, 32);

  float k2 = kvx * kvx + kvy * kvy + kvz * kvz;
  float term = 0.0f;
  if (lane < 16 && kidx < NK && k2 > 0.0f)
    term = __expf(-k2 * (1.0f / (4.0f * ALPHA_F * ALPHA_F))) / k2 *
           (sre * sre + sim * sim);
  term += __shfl_xor(term, 8, 32);
  term += __shfl_xor(term, 4, 32);
  term += __shfl_xor(term, 2, 32);
  term += __shfl_xor(term, 1, 32);
  if (lane == 0) {
    float V = bx * by * bz;
    atomicAdd(out, PREF_F * (TWO_PI_F / V) * term);
  }
}

static inline int imax(int a, int b) { return a > b ? a : b; }

extern "C" void kernel_launch(void* const* d_in, const int* in_sizes, int n_in,
                              void* d_out, int out_size, void* d_ws, size_t ws_size,
                              hipStream_t stream) {
  const float* coords  = (const float*)d_in[0];
  const float* box     = (const float*)d_in[1];
  const int*   bonds   = (const int*)  d_in[2];
  const float* b0      = (const float*)d_in[3];
  const float* kb      = (const float*)d_in[4];
  const int*   angles  = (const int*)  d_in[5];
  const float* th0     = (const float*)d_in[6];
  const float* kth     = (const float*)d_in[7];
  const float* charges = (const float*)d_in[8];
  const float* sigma   = (const float*)d_in[9];
  const float* epsilon = (const float*)d_in[10];
  const int*   types   = (const int*)  d_in[11];
  const int*   excl    = (const int*)  d_in[12];
  const int*   cexcl   = (const int*)  d_in[13];

  const int N   = in_sizes[0] / 3;
  const int nb  = in_sizes[2] / 2;
  const int na  = in_sizes[5] / 3;
  const int nx  = in_sizes[12] / 2;
  const int ncx = in_sizes[13] / 2;

  float* out   = (float*)d_out;
  int* counts  = (int*)d_ws;          /* N ints             */
  int* lists   = counts + N;          /* N * NEXCL_MAX ints */

  const int thr = 256;
  init_kernel<<<(N + thr - 1) / thr, thr, 0, stream>>>(out, counts, N);
  build_excl_kernel<<<(nx + thr - 1) / thr, thr, 0, stream>>>(excl, nx, counts, lists, N);

  int nmax = imax(imax(nb, na), imax(ncx, N));
  small_kernel<<<(nmax + thr - 1) / thr, thr, 0, stream>>>(
      coords, box, bonds, b0, kb, nb, angles, th0, kth, na, cexcl, ncx,
      charges, N, out);

  pair_kernel<<<(N + PTILE - 1) / PTILE, PTILE, 0, stream>>>(
      coords, charges, types, sigma, epsilon, box, counts, lists, N, out);

  const int kblocks = (NK + 15) / 16;  /* 307, exact */
  recip_kernel<<<kblocks, 32, 0, stream>>>(coords, charges, box, N, out);
}